// TrainableDiagonalFilter_29824252903576
// MI455X (gfx1250) — compile-verified
//
#include <hip/hip_runtime.h>

// out[row, col] = filt[row] * x[row, col]
// x: 8192 x 4096 fp32 (134 MB), filt: 8192 fp32, out: 8192 x 4096 fp32.
// Pure streaming op (0.125 FLOP/byte) -> HBM-bound, floor ~11.5 us @ 23.3 TB/s.
// Strategy: CDNA5 async global->LDS staging (GLOBAL_LOAD_ASYNC_TO_LDS_B128,
// ASYNCcnt) + s_wait_asynccnt, then scale out of LDS and write back with
// non-temporal b128 stores (268 MB streamed once -> don't thrash 192MB L2).

typedef float v4f __attribute__((ext_vector_type(4)));
typedef int   v4i __attribute__((ext_vector_type(4)));

#ifndef __has_builtin
#define __has_builtin(x) 0
#endif

#define TPB    256            // 8 waves of 32 on CDNA5
#define COLS   4096
#define ROW_F4 (COLS / 4)     // 1024 float4 per row
#define ITERS  (ROW_F4 / TPB) // 4 float4 per thread

__global__ __launch_bounds__(TPB) void diag_scale_kernel(
    const float* __restrict__ xf,
    const float* __restrict__ filt,
    float* __restrict__ outf)
{
    const int row = blockIdx.x;                // one row per block
    const float s = filt[row];                 // block-uniform -> scalar load

    const v4f* __restrict__ x4 = (const v4f*)xf + (size_t)row * ROW_F4;
    v4f*       __restrict__ o4 = (v4f*)outf + (size_t)row * ROW_F4;

#if __has_builtin(__builtin_amdgcn_global_load_async_to_lds_b128)
    // CDNA5 async path: stage the row into LDS (ASYNCcnt-tracked), then scale
    // out of LDS. Each lane reads back only its own staged 16B per iteration,
    // so s_wait_asynccnt alone is sufficient (no workgroup barrier needed).
    __shared__ v4f tile[ROW_F4];               // 16 KB of the 320 KB/WGP LDS
    #pragma unroll
    for (int k = 0; k < ITERS; ++k) {
        const int i = (int)threadIdx.x + k * TPB;
        __builtin_amdgcn_global_load_async_to_lds_b128(
            (__attribute__((address_space(1))) v4i*)(x4 + i),
            (__attribute__((address_space(3))) v4i*)(tile + i),
            /*imm offset*/0, /*cpol*/0);
    }
#if __has_builtin(__builtin_amdgcn_s_wait_asynccnt)
    __builtin_amdgcn_s_wait_asynccnt(0);
#else
    asm volatile("s_wait_asynccnt 0" ::: "memory");
#endif
    #pragma unroll
    for (int k = 0; k < ITERS; ++k) {
        const int i = (int)threadIdx.x + k * TPB;
        v4f v = tile[i];
        v *= s;
        __builtin_nontemporal_store(v, o4 + i);   // global_store_b128 th:NT
    }
#else
    // Fallback: direct streaming with non-temporal b128 loads/stores.
    #pragma unroll
    for (int k = 0; k < ITERS; ++k) {
        const int i = (int)threadIdx.x + k * TPB;
        v4f v = __builtin_nontemporal_load(x4 + i);  // global_load_b128 th:NT
        v *= s;
        __builtin_nontemporal_store(v, o4 + i);      // global_store_b128 th:NT
    }
#endif
}

extern "C" void kernel_launch(void* const* d_in, const int* in_sizes, int n_in,
                              void* d_out, int out_size, void* d_ws, size_t ws_size,
                              hipStream_t stream)
{
    const float* x    = (const float*)d_in[0];   // (8192, 4096) fp32
    const float* filt = (const float*)d_in[1];   // (8192,) fp32
    float*       out  = (float*)d_out;

    const int rows = in_sizes[1];                // 8192
    diag_scale_kernel<<<dim3(rows), dim3(TPB), 0, stream>>>(x, filt, out);
}